// GraphConv_15590731285058
// MI455X (gfx1250) — compile-verified
//
#include <hip/hip_runtime.h>
#include <cstdint>

typedef float v2f __attribute__((ext_vector_type(2)));
typedef float v8f __attribute__((ext_vector_type(8)));

#define D 128
#define LDSPITCH 132   // 132 mod 64 banks = 4 -> conflict-free b64 fragment loads

// ---------------------------------------------------------------- zero scratch
__global__ void zero_ws(float* __restrict__ p, long long n) {
    long long i = (long long)blockIdx.x * blockDim.x + threadIdx.x;
    long long stride = (long long)gridDim.x * blockDim.x;
    for (; i < n; i += stride) p[i] = 0.0f;
}

// ---------------------------------------------------------------- degree count
__global__ void degrees_kernel(const int* __restrict__ ei, int E,
                               float* __restrict__ degOut, float* __restrict__ degIn) {
    int e = blockIdx.x * blockDim.x + threadIdx.x;
    if (e >= E) return;
    int s = ei[e];
    int d = ei[E + e];
    __hip_atomic_fetch_add(&degOut[s], 1.0f, __ATOMIC_RELAXED, __HIP_MEMORY_SCOPE_AGENT);
    __hip_atomic_fetch_add(&degIn[d],  1.0f, __ATOMIC_RELAXED, __HIP_MEMORY_SCOPE_AGENT);
}

// ------------------------------------------------- edge scatter-add aggregation
// 32 threads per edge, 4 floats per thread (float4 gather + 4 f32 atomics).
// feat and agg are L2-resident (25.6 MB each vs 192 MB L2).
__global__ void aggregate_kernel(const int* __restrict__ ei, int E,
                                 const float* __restrict__ feat,
                                 const float* __restrict__ degOut,
                                 float* __restrict__ agg) {
    long long t = (long long)blockIdx.x * blockDim.x + threadIdx.x;
    int e = (int)(t >> 5);
    if (e >= E) return;
    int q = ((int)t & 31) * 4;
    int s = ei[e];
    int d = ei[E + e];
    float sc = rsqrtf(fmaxf(degOut[s], 1.0f));
    const float4 v = *(const float4*)(feat + (long long)s * D + q);
    float* dst = agg + (long long)d * D + q;
    __hip_atomic_fetch_add(dst + 0, v.x * sc, __ATOMIC_RELAXED, __HIP_MEMORY_SCOPE_AGENT);
    __hip_atomic_fetch_add(dst + 1, v.y * sc, __ATOMIC_RELAXED, __HIP_MEMORY_SCOPE_AGENT);
    __hip_atomic_fetch_add(dst + 2, v.z * sc, __ATOMIC_RELAXED, __HIP_MEMORY_SCOPE_AGENT);
    __hip_atomic_fetch_add(dst + 3, v.w * sc, __ATOMIC_RELAXED, __HIP_MEMORY_SCOPE_AGENT);
}

// -------------------------------------- fused WMMA GEMM + scale + residual + LN
// out = relu(LayerNorm((agg @ W^T + b) * rsqrt(deg_in) + feat) * gamma + beta)
// One workgroup = 8 waves = 128 rows. Each wave: 16 rows x 128 cols, 8 C-tiles,
// 32 k-steps of v_wmma_f32_16x16x4_f32 per tile column.
__global__ __launch_bounds__(256)
void gemm_ln_kernel(const float* __restrict__ agg,
                    const float* __restrict__ Wmat,
                    const float* __restrict__ bias,
                    const float* __restrict__ gamma,
                    const float* __restrict__ beta,
                    const float* __restrict__ feat,
                    const float* __restrict__ degIn,
                    float* __restrict__ out, int N) {
    __shared__ float wlds[D * LDSPITCH];

    const int tid = threadIdx.x;

    // Stage W into LDS (row-major, padded pitch). 256 threads x 16 float4 each.
    {
        int row  = tid >> 1;            // 0..127
        int half = (tid & 1) * 64;      // 0 or 64
        const float4* src = (const float4*)(Wmat + row * D + half);
        float4*       dst = (float4*)(&wlds[row * LDSPITCH + half]);
#pragma unroll
        for (int i = 0; i < 16; ++i) dst[i] = src[i];
    }
    __syncthreads();

    const int wave  = tid >> 5;
    const int lane  = tid & 31;
    const int ln16  = lane & 15;
    const int khalf = lane >> 4;        // 0: K pair {0,1}, 1: K pair {2,3}
    const int rowBase = blockIdx.x * 128 + wave * 16;
    const int arow = min(rowBase + ln16, N - 1);   // clamp: EXEC stays all-1s

    v8f acc[8];
#pragma unroll
    for (int j = 0; j < 8; ++j) acc[j] = (v8f){0, 0, 0, 0, 0, 0, 0, 0};

    // A fragment source: agg[arow, k0 + 2*khalf .. +1]  (ISA 16x4 f32 A layout)
    const float* aptr = agg + (long long)arow * D + 2 * khalf;

#pragma unroll 4
    for (int k0 = 0; k0 < D; k0 += 4) {
        v2f a = *(const v2f*)(aptr + k0);
#pragma unroll
        for (int j = 0; j < 8; ++j) {
            // B[k, n] = W[n, k]; fragment: lane -> W[16j+ln16, k0+2*khalf +{0,1}]
            v2f b = *(const v2f*)(&wlds[(j * 16 + ln16) * LDSPITCH + k0 + 2 * khalf]);
            acc[j] = __builtin_amdgcn_wmma_f32_16x16x4_f32(
                false, a, false, b, (short)0, acc[j], false, false);
        }
    }

    // Per-lane column-indexed params: col = 16*j + ln16
    float bcol[8], gcol[8], btcol[8];
#pragma unroll
    for (int j = 0; j < 8; ++j) {
        int col = j * 16 + ln16;
        bcol[j]  = bias[col];
        gcol[j]  = gamma[col];
        btcol[j] = beta[col];
    }

    // Epilogue. C/D layout: VGPR v, lane L -> row = v + 8*(L>>4), col = 16j + (L&15).
    // Each row's 128 values live in one 16-lane half -> butterfly reduce width 16.
#pragma unroll
    for (int v = 0; v < 8; ++v) {
        int row  = rowBase + v + 8 * khalf;
        int rowc = min(row, N - 1);
        float rs = rsqrtf(fmaxf(degIn[rowc], 1.0f));
        const float* frow = feat + (long long)rowc * D;

        float val[8];
        float sum = 0.0f, sumsq = 0.0f;
#pragma unroll
        for (int j = 0; j < 8; ++j) {
            int col = j * 16 + ln16;
            float x = (acc[j][v] + bcol[j]) * rs + frow[col];
            val[j] = x;
            sum   += x;
            sumsq += x * x;
        }
#pragma unroll
        for (int m = 1; m < 16; m <<= 1) {
            sum   += __shfl_xor(sum,   m, 32);
            sumsq += __shfl_xor(sumsq, m, 32);
        }
        float mean = sum * (1.0f / D);
        float var  = sumsq * (1.0f / D) - mean * mean;
        float inv  = rsqrtf(var + 1e-5f);

        if (row < N) {
            float* orow = out + (long long)row * D;
#pragma unroll
            for (int j = 0; j < 8; ++j) {
                int col = j * 16 + ln16;
                float y = (val[j] - mean) * inv * gcol[j] + btcol[j];
                orow[col] = fmaxf(y, 0.0f);
            }
        }
    }
}

// ------------------------------------------------------------------- launcher
extern "C" void kernel_launch(void* const* d_in, const int* in_sizes, int n_in,
                              void* d_out, int out_size, void* d_ws, size_t ws_size,
                              hipStream_t stream) {
    const float* feat  = (const float*)d_in[0];
    const int*   ei    = (const int*)d_in[1];     // [2, E] int32 (src row, then dst row)
    const float* Wmat  = (const float*)d_in[2];
    const float* bias  = (const float*)d_in[3];
    const float* gamma = (const float*)d_in[4];
    const float* beta  = (const float*)d_in[5];

    const int N = in_sizes[0] / D;
    const int E = in_sizes[1] / 2;

    // Workspace layout: degOut[N] | degIn[N] | (64-float aligned) agg[N*D]
    float* ws      = (float*)d_ws;
    float* degOut  = ws;
    float* degIn   = ws + N;
    size_t aggOff  = ((size_t)(2 * N) + 63) & ~(size_t)63;
    float* agg     = ws + aggOff;
    long long zn   = (long long)aggOff + (long long)N * D;

    zero_ws<<<2048, 256, 0, stream>>>(ws, zn);

    degrees_kernel<<<(E + 255) / 256, 256, 0, stream>>>(ei, E, degOut, degIn);

    long long aggThreads = (long long)E * 32;
    aggregate_kernel<<<(int)((aggThreads + 255) / 256), 256, 0, stream>>>(
        ei, E, feat, degOut, agg);

    gemm_ln_kernel<<<(N + 127) / 128, 256, 0, stream>>>(
        agg, Wmat, bias, gamma, beta, feat, degIn, (float*)d_out, N);
}